// BCE_topK_loss_sep_channel_1915555414719
// MI455X (gfx1250) — compile-verified
//
#include <hip/hip_runtime.h>

// BCE-with-logits + per-row top-k% mean for [B=2,C=14,S=128^3] f32.
// Exact 2-sweep radix(16+16) selection; LDS-privatized 64K-bin histograms
// (CDNA5 320KB LDS/WGP); async global->LDS double-buffered staging in sweep 2.

#define NROWS 28
#define W1 16          // histogram blocks per row (sweep 1)
#define W2 64          // blocks per row (sweep 2)
#define NBINS 65536
#define HIST_BYTES ((size_t)NROWS * NBINS * 4)

typedef float v4f __attribute__((ext_vector_type(4)));

__device__ __forceinline__ float bce_loss(float x, float t) {
    // max(x,0) - x*t + log1p(exp(-|x|)) ; always > 0 -> bit pattern is order-preserving
    return fmaxf(x, 0.0f) - x * t + log1pf(expf(-fabsf(x)));
}

// ---- CDNA5 async global->LDS (ASYNCcnt path). Low 32 bits of a flat shared
// pointer are the workgroup-relative LDS byte offset on gfx1250.
__device__ __forceinline__ void async_ld16(unsigned lds_off, const void* gaddr) {
    asm volatile("global_load_async_to_lds_b128 %0, %1, off"
                 :: "v"(lds_off), "v"((unsigned long long)gaddr)
                 : "memory");
}
__device__ __forceinline__ void wait_async_le2() {
    asm volatile("s_wait_asynccnt 0x2" ::: "memory");
}
__device__ __forceinline__ void wait_async_0() {
    asm volatile("s_wait_asynccnt 0x0" ::: "memory");
}
__device__ __forceinline__ void wait_ds_0() {
    asm volatile("s_wait_dscnt 0x0" ::: "memory");
}

// ---------------- Sweep 1: per-row 64K-bin histogram of loss bits[31:16] ----
__global__ __launch_bounds__(1024)
void hist1_kernel(const float* __restrict__ X, const float* __restrict__ T,
                  unsigned* __restrict__ ghist, long long S) {
    extern __shared__ unsigned lhist[];            // 65536 u32 = 256 KB dynamic LDS
    const int tid = threadIdx.x;
    const int row = blockIdx.x / W1;
    const int sub = blockIdx.x % W1;
    for (int i = tid; i < NBINS; i += 1024) lhist[i] = 0u;
    __syncthreads();

    const long long chunk = S / W1;                // S=2^21 -> 131072
    const long long base  = (long long)row * S + (long long)sub * chunk;
    const v4f* x4 = (const v4f*)(X + base);
    const v4f* t4 = (const v4f*)(T + base);
    const int iters = (int)(chunk / (4 * 1024));   // 32
    for (int i = 0; i < iters; ++i) {
        v4f xv = __builtin_nontemporal_load(&x4[(long long)i * 1024 + tid]);
        v4f tv = __builtin_nontemporal_load(&t4[(long long)i * 1024 + tid]);
#pragma unroll
        for (int j = 0; j < 4; ++j) {
            float z = bce_loss(xv[j], tv[j]);
            atomicAdd(&lhist[__float_as_uint(z) >> 16], 1u);
        }
    }
    __syncthreads();
    unsigned* gh = ghist + (size_t)row * NBINS;
    for (int i = tid; i < NBINS; i += 1024) {
        unsigned c = lhist[i];
        if (c) atomicAdd(&gh[i], c);
    }
}

// ---------------- Scan 1: find threshold bin b0 and residual r0 per row -----
__global__ __launch_bounds__(1024)
void scan1_kernel(const unsigned* __restrict__ ghist, const int* __restrict__ kp,
                  unsigned* __restrict__ params, long long S) {
    __shared__ unsigned csum[1024];
    const int row = blockIdx.x;
    const int tid = threadIdx.x;
    const unsigned* h = ghist + (size_t)row * NBINS;
    unsigned s = 0;
    const int base = tid * 64;
    for (int i = 0; i < 64; ++i) s += h[base + i];
    csum[tid] = s;
    __syncthreads();
    if (tid == 0) {
        double v = (double)S * (double)(*kp) / 100.0;
        long long n = __double2ll_rn(v);           // RNE == Python round()
        if (n < 1) n = 1;
        if (n > S) n = S;
        unsigned need = (unsigned)n;
        unsigned cum = 0;
        int c = 1023;
        for (; c >= 0; --c) {
            if (cum + csum[c] >= need) break;
            cum += csum[c];
        }
        if (c < 0) c = 0;
        unsigned b0 = (unsigned)(c * 64);
        unsigned r0 = (need > cum) ? (need - cum) : 0u;
        for (int bb = 63; bb >= 0; --bb) {
            unsigned cc = h[c * 64 + bb];
            if (cum + cc >= need) { b0 = (unsigned)(c * 64 + bb); r0 = need - cum; break; }
            cum += cc;
        }
        params[row * 2 + 0] = b0;
        params[row * 2 + 1] = r0;
        if (row == 0) params[NROWS * 2] = (unsigned)n;
    }
}

// ---------------- Sweep 2: exact sum above b0 + low-16 hist inside b0 -------
__global__ __launch_bounds__(256)
void pass2_kernel(const float* __restrict__ X, const float* __restrict__ T,
                  const unsigned* __restrict__ params,
                  unsigned* __restrict__ ghist2,
                  double* __restrict__ blockPartials, long long S) {
    __shared__ float bx[2][1024];
    __shared__ float bt[2][1024];
    __shared__ double red[256];
    const int tid = threadIdx.x;
    const int row = blockIdx.x / W2;
    const int sub = blockIdx.x % W2;
    const long long chunk = S / W2;                // 32768
    const long long base  = (long long)row * S + (long long)sub * chunk;
    const unsigned b0 = params[row * 2 + 0];
    unsigned* __restrict__ h2 = ghist2 + (size_t)row * NBINS;
    const float* xc = X + base;
    const float* tc = T + base;
    const int iters = (int)(chunk / 1024);         // 32

    unsigned lx[2], lt[2];
    lx[0] = (unsigned)(unsigned long long)&bx[0][tid * 4];
    lx[1] = (unsigned)(unsigned long long)&bx[1][tid * 4];
    lt[0] = (unsigned)(unsigned long long)&bt[0][tid * 4];
    lt[1] = (unsigned)(unsigned long long)&bt[1][tid * 4];

    double acc = 0.0;
    // prologue: stage iter 0 into buffer 0 (each lane owns slot tid -> no barriers)
    wait_ds_0();
    async_ld16(lx[0], xc + tid * 4);
    async_ld16(lt[0], tc + tid * 4);
    for (int it = 0; it < iters; ++it) {
        const int cur = it & 1;
        if (it + 1 < iters) {
            wait_ds_0();  // our prior LDS reads of buffer cur^1 must land before overwrite
            async_ld16(lx[cur ^ 1], xc + (long long)(it + 1) * 1024 + tid * 4);
            async_ld16(lt[cur ^ 1], tc + (long long)(it + 1) * 1024 + tid * 4);
            wait_async_le2();   // in-order completion => oldest pair (cur) is done
        } else {
            wait_async_0();
        }
#pragma unroll
        for (int j = 0; j < 4; ++j) {
            float x = bx[cur][tid * 4 + j];
            float t = bt[cur][tid * 4 + j];
            float z = bce_loss(x, t);
            unsigned key = __float_as_uint(z);
            unsigned hi = key >> 16;
            if (hi > b0)        acc += (double)z;                     // fully included
            else if (hi == b0)  atomicAdd(&h2[key & 0xffffu], 1u);    // boundary bin
        }
    }
    red[tid] = acc;
    __syncthreads();
    for (int s = 128; s > 0; s >>= 1) {            // fixed-order tree => deterministic
        if (tid < s) red[tid] += red[tid + s];
        __syncthreads();
    }
    if (tid == 0) blockPartials[blockIdx.x] = red[0];
}

// ---------------- Scan 2: exact boundary-bin contribution per row -----------
__global__ __launch_bounds__(256)
void scan2_kernel(const unsigned* __restrict__ ghist2, const unsigned* __restrict__ params,
                  double* __restrict__ rowPartials) {
    __shared__ double wsum[256];
    __shared__ unsigned csum[256];
    const int row = blockIdx.x;
    const int tid = threadIdx.x;
    const unsigned b0 = params[row * 2 + 0];
    const unsigned r0 = params[row * 2 + 1];
    const unsigned hi = b0 << 16;
    const unsigned* h = ghist2 + (size_t)row * NBINS;
    double w = 0.0;
    unsigned cs = 0;
    const int base = tid * 256;
    for (int i = 0; i < 256; ++i) {
        unsigned c = h[base + i];
        if (c) {
            cs += c;
            w += (double)c * (double)__uint_as_float(hi | (unsigned)(base + i));
        }
    }
    wsum[tid] = w; csum[tid] = cs;
    __syncthreads();
    if (tid == 0) {
        double tot = 0.0;
        unsigned rem = r0;
        for (int c = 255; c >= 0 && rem; --c) {
            if (csum[c] <= rem) { tot += wsum[c]; rem -= csum[c]; }
            else {
                for (int bb = 255; bb >= 0 && rem; --bb) {
                    unsigned cc = h[c * 256 + bb];
                    unsigned m = (cc < rem) ? cc : rem;
                    if (m) tot += (double)m * (double)__uint_as_float(hi | (unsigned)(c * 256 + bb));
                    rem -= m;
                }
                break;
            }
        }
        rowPartials[row] = tot;
    }
}

// ---------------- Finalize: deterministic serial combine + mean -------------
__global__ void finalize_kernel(const double* __restrict__ blockPartials, int nBlocks,
                                const double* __restrict__ rowPartials,
                                const unsigned* __restrict__ params,
                                float* __restrict__ out) {
    if (threadIdx.x == 0 && blockIdx.x == 0) {
        double t = 0.0;
        for (int i = 0; i < nBlocks; ++i) t += blockPartials[i];
        for (int i = 0; i < NROWS; ++i)  t += rowPartials[i];
        unsigned n = params[NROWS * 2];
        out[0] = (float)(t / ((double)NROWS * (double)n));
    }
}

extern "C" void kernel_launch(void* const* d_in, const int* in_sizes, int n_in,
                              void* d_out, int out_size, void* d_ws, size_t ws_size,
                              hipStream_t stream) {
    (void)n_in; (void)out_size; (void)ws_size;
    const float* X  = (const float*)d_in[0];
    const float* T  = (const float*)d_in[1];
    const int*   kp = (const int*)d_in[2];
    float* out = (float*)d_out;

    const long long total = (long long)in_sizes[0];
    const long long S = total / NROWS;             // 2^21; divisible by W1*4096 and W2*1024

    unsigned char* ws = (unsigned char*)d_ws;
    unsigned* ghist        = (unsigned*)ws;                                   // 7,340,032 B
    unsigned* params       = (unsigned*)(ws + HIST_BYTES);                    // 2*28+1 u32
    double*   blockPartial = (double*)(ws + HIST_BYTES + 256);                // 1792 doubles
    const int nBlocks2 = NROWS * W2;
    double*   rowPartial   = (double*)(ws + HIST_BYTES + 256 + (size_t)nBlocks2 * 8);

    // allow 256KB dynamic LDS for the 64K-bin histogram (CDNA5: 320KB/WGP)
    hipFuncSetAttribute((const void*)hist1_kernel,
                        hipFuncAttributeMaxDynamicSharedMemorySize, NBINS * 4);

    hipMemsetAsync(ghist, 0, HIST_BYTES, stream);
    hist1_kernel<<<dim3(NROWS * W1), dim3(1024), NBINS * 4, stream>>>(X, T, ghist, S);
    scan1_kernel<<<dim3(NROWS), dim3(1024), 0, stream>>>(ghist, kp, params, S);
    hipMemsetAsync(ghist, 0, HIST_BYTES, stream);  // reuse as level-2 histogram
    pass2_kernel<<<dim3(nBlocks2), dim3(256), 0, stream>>>(X, T, params, ghist, blockPartial, S);
    scan2_kernel<<<dim3(NROWS), dim3(256), 0, stream>>>(ghist, params, rowPartial);
    finalize_kernel<<<dim3(1), dim3(1), 0, stream>>>(blockPartial, nBlocks2, rowPartial, params, out);
}